// StaticKVForward_20512763806101
// MI455X (gfx1250) — compile-verified
//
#include <hip/hip_runtime.h>

// ---------------- model constants (from reference) ----------------
#define Lc    2
#define Bc    4
#define Dm    2048
#define NH    16
#define NKV   8
#define HDc   128
#define FF    8192
#define VOC   32000
#define SMAX  4096
#define EPSc  1e-6f
#define SCALEc 0.08838834764831845f   // 1/sqrt(128)
#define MPAD  16                       // WMMA M tile (B=4 padded with zeros)
#define PV_CHUNKS 8

typedef __attribute__((ext_vector_type(2))) float v2f;
typedef __attribute__((ext_vector_type(4))) float v4f;
typedef __attribute__((ext_vector_type(8))) float v8f;

// ---------------- utility kernels ----------------
__global__ __launch_bounds__(256)
void zero_kernel(float* __restrict__ p, int n) {
  int i = blockIdx.x * 256 + threadIdx.x;
  if (i < n) p[i] = 0.f;
}

__global__ __launch_bounds__(256)
void embed_kernel(const int* __restrict__ tok, const float* __restrict__ ew,
                  float* __restrict__ X) {
  int b = blockIdx.y;
  int d = blockIdx.x * 256 + threadIdx.x;
  X[(size_t)b * Dm + d] = ew[(size_t)tok[b] * Dm + d];
}

__global__ __launch_bounds__(256)
void rmsnorm_kernel(const float* __restrict__ X, const float* __restrict__ w,
                    float* __restrict__ Y) {
  int b = blockIdx.x;                       // rows 0..Bc-1 (pad rows stay zero)
  const float* x = X + (size_t)b * Dm;
  float*       y = Y + (size_t)b * Dm;
  __shared__ float red[8];
  __shared__ float bc;
  float s = 0.f;
  for (int d = threadIdx.x; d < Dm; d += 256) { float v = x[d]; s += v * v; }
  for (int o = 16; o; o >>= 1) s += __shfl_down(s, o, 32);   // wave32
  if ((threadIdx.x & 31) == 0) red[threadIdx.x >> 5] = s;
  __syncthreads();
  if (threadIdx.x == 0) {
    float t = 0.f;
    for (int i = 0; i < 8; ++i) t += red[i];
    bc = rsqrtf(t / (float)Dm + EPSc);
  }
  __syncthreads();
  float r = bc;
  for (int d = threadIdx.x; d < Dm; d += 256) y[d] = x[d] * r * w[d];
}

// ---------------- fp32 WMMA GEMM: C[16 x N] = A[16 x K] @ W[K x N] (+Res) ----
// 8 waves/block, one 16x16 output tile per wave. K-loop: outer step of 32 with
// one in-bounds global_prefetch per body, inner 8x unrolled steps of 4 doing
// V_WMMA_F32_16X16X4_F32 on pointer-stepped non-temporal weight loads.
// Requires K % 32 == 0 (holds for all uses: 2048, 8192).
__global__ __launch_bounds__(256)
void wmma_gemm_kernel(const float* __restrict__ A, const float* __restrict__ W,
                      float* __restrict__ C, const float* __restrict__ Res,
                      int K, int N, int ldc, int m_real) {
  const int lane  = threadIdx.x & 31;
  const int tile  = blockIdx.x * 8 + (threadIdx.x >> 5);
  const int ncol0 = tile * 16;
  if (ncol0 >= N) return;                       // wave-uniform
  const int m     = lane & 15;
  const int khalf = (lane >> 4) << 1;           // 0 or 2
  const int ncol  = ncol0 + m;

  // A 16x4 frag: lanes0-15 hold K=kb,kb+1 of row m; lanes16-31 hold kb+2,kb+3.
  // (m*K + khalf + kb) is always even -> 8B aligned -> single b64 load.
  const float* aRow = A + (size_t)m * K + khalf;
  // B 4x16 frag: lane half picks the K pair, lane&15 is the column.
  const float* wP0 = W + (size_t)khalf * N + ncol;
  const float* wP1 = wP0 + N;
  const size_t wStep = (size_t)4 * N;

  v8f acc = {};
  const int kbEnd = K - 32;
  int kb = 0;
  for (; kb < kbEnd; kb += 32) {
    // prefetch 32 K-rows ahead (always within W for kb < K-32)
    __builtin_prefetch((const void*)(wP0 + 8 * wStep), 0, 3);
#pragma unroll
    for (int u = 0; u < 8; ++u) {
      v2f a = *(const v2f*)(aRow + kb + 4 * u);
      v2f b;
      b.x = __builtin_nontemporal_load(wP0);
      b.y = __builtin_nontemporal_load(wP1);
      wP0 += wStep;
      wP1 += wStep;
      acc = __builtin_amdgcn_wmma_f32_16x16x4_f32(false, a, false, b,
                                                  (short)0, acc, false, false);
    }
  }
  // tail 32 K (no prefetch)
#pragma unroll
  for (int u = 0; u < 8; ++u) {
    v2f a = *(const v2f*)(aRow + kb + 4 * u);
    v2f b;
    b.x = __builtin_nontemporal_load(wP0);
    b.y = __builtin_nontemporal_load(wP1);
    wP0 += wStep;
    wP1 += wStep;
    acc = __builtin_amdgcn_wmma_f32_16x16x4_f32(false, a, false, b,
                                                (short)0, acc, false, false);
  }

  // C/D layout: vgpr r -> row r + 8*(lane>=16), col = ncol
  const int rbase = (lane >> 4) * 8;
#pragma unroll
  for (int r = 0; r < 8; ++r) {
    int row = rbase + r;
    if (row < m_real) {
      size_t off = (size_t)row * ldc + ncol;
      float v = acc[r];
      if (Res) v += Res[off];
      C[off] = v;
    }
  }
}

// ---------------- per-head RMSNorm + RoPE on q and new k (in place) ---------
__global__ __launch_bounds__(HDc)
void qknorm_rope_kernel(float* __restrict__ Q, float* __restrict__ Kn,
                        const float* __restrict__ qw, const float* __restrict__ kw,
                        const int* __restrict__ cpos) {
  int hh = blockIdx.x;                      // [0,Bc*NH): q heads, then k heads
  int t  = threadIdx.x;
  float* ptr; const float* w;
  if (hh < Bc * NH) {
    int b = hh / NH, h = hh % NH;
    ptr = Q + ((size_t)b * NH + h) * HDc;  w = qw;
  } else {
    int i2 = hh - Bc * NH;
    int b = i2 / NKV, k = i2 % NKV;
    ptr = Kn + ((size_t)b * NKV + k) * HDc; w = kw;
  }
  __shared__ float sh[HDc];
  __shared__ float red[4];
  __shared__ float rr;
  float v = ptr[t];
  float s = v * v;
  for (int o = 16; o; o >>= 1) s += __shfl_down(s, o, 32);
  if ((t & 31) == 0) red[t >> 5] = s;
  __syncthreads();
  if (t == 0) rr = rsqrtf((red[0] + red[1] + red[2] + red[3]) / (float)HDc + EPSc);
  __syncthreads();
  sh[t] = v * rr * w[t];
  __syncthreads();
  if (t < HDc / 2) {
    float x1 = sh[t], x2 = sh[t + HDc / 2];
    float inv = powf(1.0e6f, -(float)t / (float)(HDc / 2));
    float ang = (float)cpos[0] * inv;
    float c = cosf(ang), sn = sinf(ang);
    ptr[t]           = x1 * c - x2 * sn;
    ptr[t + HDc / 2] = x2 * c + x1 * sn;
  }
}

// ---------------- attention scores (masked; new k substituted at cp) --------
__global__ __launch_bounds__(256)
void attn_scores_kernel(const float* __restrict__ Q, const float* __restrict__ Kn,
                        const float* __restrict__ kcache, float* __restrict__ S,
                        const int* __restrict__ cpos) {
  int bh = blockIdx.x;
  int b = bh / NH, h = bh % NH;
  int kv = h / (NH / NKV);
  int pos = blockIdx.y * 256 + threadIdx.x;
  int cp = cpos[0];
  float sc = -1e9f;
  if (pos <= cp) {
    const v4f* qr = (const v4f*)(Q + ((size_t)b * NH + h) * HDc);
    const v4f* kr = (pos == cp)
        ? (const v4f*)(Kn + ((size_t)b * NKV + kv) * HDc)
        : (const v4f*)(kcache + (((size_t)b * NKV + kv) * SMAX + pos) * HDc);
    float acc = 0.f;
#pragma unroll
    for (int i = 0; i < HDc / 4; ++i) {
      v4f qv = qr[i];
      v4f kx = __builtin_nontemporal_load(kr + i);   // cache is read-once
      acc += qv.x * kx.x + qv.y * kx.y + qv.z * kx.z + qv.w * kx.w;
    }
    sc = acc * SCALEc;
  }
  S[(size_t)bh * SMAX + pos] = sc;
}

__global__ __launch_bounds__(256)
void softmax_kernel(float* __restrict__ S) {
  float* s = S + (size_t)blockIdx.x * SMAX;
  __shared__ float red[8];
  __shared__ float bc;
  float m = -3.4e38f;
  for (int p = threadIdx.x; p < SMAX; p += 256) m = fmaxf(m, s[p]);
  for (int o = 16; o; o >>= 1) m = fmaxf(m, __shfl_down(m, o, 32));
  if ((threadIdx.x & 31) == 0) red[threadIdx.x >> 5] = m;
  __syncthreads();
  if (threadIdx.x == 0) {
    float t = red[0];
    for (int i = 1; i < 8; ++i) t = fmaxf(t, red[i]);
    bc = t;
  }
  __syncthreads();
  m = bc;
  __syncthreads();
  float sum = 0.f;
  for (int p = threadIdx.x; p < SMAX; p += 256) {
    float e = expf(s[p] - m);
    s[p] = e;
    sum += e;
  }
  for (int o = 16; o; o >>= 1) sum += __shfl_down(sum, o, 32);
  if ((threadIdx.x & 31) == 0) red[threadIdx.x >> 5] = sum;
  __syncthreads();
  if (threadIdx.x == 0) {
    float t = 0.f;
    for (int i = 0; i < 8; ++i) t += red[i];
    bc = 1.f / t;
  }
  __syncthreads();
  float inv = bc;
  for (int p = threadIdx.x; p < SMAX; p += 256) s[p] *= inv;
}

// ---------------- P@V, chunked partials (deterministic, no atomics) ---------
__global__ __launch_bounds__(HDc)
void attn_pv_kernel(const float* __restrict__ P, const float* __restrict__ Vn,
                    const float* __restrict__ vcache, float* __restrict__ Opart,
                    const int* __restrict__ cpos) {
  int bh = blockIdx.x;
  int b = bh / NH, h = bh % NH;
  int kv = h / (NH / NKV);
  int d = threadIdx.x;
  int cp = cpos[0];
  const int span = SMAX / PV_CHUNKS;
  int p0 = blockIdx.y * span;
  int p1 = p0 + span;
  if (p1 > cp + 1) p1 = cp + 1;
  const float* prow  = P + (size_t)bh * SMAX;
  const float* vbase = vcache + (((size_t)b * NKV + kv) * SMAX) * HDc;
  const float* vnew  = Vn + ((size_t)b * NKV + kv) * HDc;
  float acc = 0.f;
  const float* vr = vbase + (size_t)p0 * HDc + d;
  for (int pos = p0; pos < p1; ++pos) {
    float vv = (pos == cp) ? vnew[d] : __builtin_nontemporal_load(vr);
    acc += prow[pos] * vv;
    vr += HDc;
  }
  Opart[((size_t)blockIdx.y * (Bc * NH) + bh) * HDc + d] = acc;
}

__global__ __launch_bounds__(256)
void reduce_o_kernel(const float* __restrict__ Opart, float* __restrict__ O) {
  int i = blockIdx.x * 256 + threadIdx.x;   // i < Bc*NH*HDc; NH*HDc == Dm
  float s = 0.f;
#pragma unroll
  for (int c = 0; c < PV_CHUNKS; ++c)
    s += Opart[(size_t)c * (Bc * NH * HDc) + i];
  O[i] = s;                                  // rows 0..Bc-1 of padded [16,Dm]
}

__global__ __launch_bounds__(256)
void silu_mul_kernel(float* __restrict__ G, const float* __restrict__ U) {
  int i = blockIdx.x * 256 + threadIdx.x;   // covers rows 0..Bc-1 only
  float g = G[i];
  G[i] = g / (1.f + expf(-g)) * U[i];
}

// ---------------- host orchestration ----------------
extern "C" void kernel_launch(void* const* d_in, const int* in_sizes, int n_in,
                              void* d_out, int out_size, void* d_ws, size_t ws_size,
                              hipStream_t stream) {
  const int*   tok    = (const int*)d_in[0];
  const int*   cpos   = (const int*)d_in[1];
  const float* kcache = (const float*)d_in[2];
  const float* vcache = (const float*)d_in[3];
  const float* ew     = (const float*)d_in[4];
  const float* ln1    = (const float*)d_in[5];
  const float* wq     = (const float*)d_in[6];
  const float* wk     = (const float*)d_in[7];
  const float* wv     = (const float*)d_in[8];
  const float* qnw    = (const float*)d_in[9];
  const float* knw    = (const float*)d_in[10];
  const float* wo     = (const float*)d_in[11];
  const float* ln2    = (const float*)d_in[12];
  const float* wg     = (const float*)d_in[13];
  const float* wu     = (const float*)d_in[14];
  const float* wd     = (const float*)d_in[15];
  const float* fnw    = (const float*)d_in[16];
  const float* lmh    = (const float*)d_in[17];
  float* out = (float*)d_out;

  float* ws = (float*)d_ws;
  size_t o = 0;
  float* X  = ws + o; o += (size_t)MPAD * Dm;          // residual [16,2048]
  float* Hb = ws + o; o += (size_t)MPAD * Dm;          // normed activations
  float* Q  = ws + o; o += (size_t)MPAD * Dm;          // q proj
  float* Kn = ws + o; o += (size_t)MPAD * (NKV * HDc); // new k
  float* Vn = ws + o; o += (size_t)MPAD * (NKV * HDc); // new v
  float* O  = ws + o; o += (size_t)MPAD * Dm;          // attn out
  float* G  = ws + o; o += (size_t)MPAD * FF;          // gate (then silu*up)
  float* U  = ws + o; o += (size_t)MPAD * FF;          // up
  float* S  = ws + o; o += (size_t)Bc * NH * SMAX;     // scores/probs
  float* Op = ws + o; o += (size_t)PV_CHUNKS * Bc * NH * HDc;
  int total = (int)o;

  zero_kernel<<<(total + 255) / 256, 256, 0, stream>>>(ws, total);
  embed_kernel<<<dim3(Dm / 256, Bc), 256, 0, stream>>>(tok, ew, X);

  for (int l = 0; l < Lc; ++l) {
    const float* wq_l = wq + (size_t)l * Dm * (NH * HDc);
    const float* wk_l = wk + (size_t)l * Dm * (NKV * HDc);
    const float* wv_l = wv + (size_t)l * Dm * (NKV * HDc);
    const float* wo_l = wo + (size_t)l * (NH * HDc) * Dm;
    const float* wg_l = wg + (size_t)l * Dm * FF;
    const float* wu_l = wu + (size_t)l * Dm * FF;
    const float* wd_l = wd + (size_t)l * FF * Dm;
    const float* kc_l = kcache + (size_t)l * Bc * NKV * SMAX * HDc;
    const float* vc_l = vcache + (size_t)l * Bc * NKV * SMAX * HDc;

    rmsnorm_kernel<<<Bc, 256, 0, stream>>>(X, ln1 + (size_t)l * Dm, Hb);
    wmma_gemm_kernel<<<(NH * HDc / 16) / 8, 256, 0, stream>>>(
        Hb, wq_l, Q, nullptr, Dm, NH * HDc, NH * HDc, MPAD);
    wmma_gemm_kernel<<<(NKV * HDc / 16) / 8, 256, 0, stream>>>(
        Hb, wk_l, Kn, nullptr, Dm, NKV * HDc, NKV * HDc, MPAD);
    wmma_gemm_kernel<<<(NKV * HDc / 16) / 8, 256, 0, stream>>>(
        Hb, wv_l, Vn, nullptr, Dm, NKV * HDc, NKV * HDc, MPAD);
    qknorm_rope_kernel<<<Bc * (NH + NKV), HDc, 0, stream>>>(
        Q, Kn, qnw + (size_t)l * HDc, knw + (size_t)l * HDc, cpos);
    attn_scores_kernel<<<dim3(Bc * NH, SMAX / 256), 256, 0, stream>>>(
        Q, Kn, kc_l, S, cpos);
    softmax_kernel<<<Bc * NH, 256, 0, stream>>>(S);
    attn_pv_kernel<<<dim3(Bc * NH, PV_CHUNKS), HDc, 0, stream>>>(
        S, Vn, vc_l, Op, cpos);
    reduce_o_kernel<<<(Bc * NH * HDc) / 256, 256, 0, stream>>>(Op, O);
    wmma_gemm_kernel<<<(Dm / 16) / 8, 256, 0, stream>>>(
        O, wo_l, X, X, NH * HDc, Dm, Dm, MPAD);
    rmsnorm_kernel<<<Bc, 256, 0, stream>>>(X, ln2 + (size_t)l * Dm, Hb);
    wmma_gemm_kernel<<<(FF / 16) / 8, 256, 0, stream>>>(
        Hb, wg_l, G, nullptr, Dm, FF, FF, MPAD);
    wmma_gemm_kernel<<<(FF / 16) / 8, 256, 0, stream>>>(
        Hb, wu_l, U, nullptr, Dm, FF, FF, MPAD);
    silu_mul_kernel<<<(Bc * FF) / 256, 256, 0, stream>>>(G, U);
    wmma_gemm_kernel<<<(Dm / 16) / 8, 256, 0, stream>>>(
        G, wd_l, X, X, FF, Dm, Dm, MPAD);
  }

  rmsnorm_kernel<<<Bc, 256, 0, stream>>>(X, fnw, Hb);
  wmma_gemm_kernel<<<(VOC / 16) / 8, 256, 0, stream>>>(
      Hb, lmh, out, nullptr, Dm, VOC, VOC, Bc);
}